// DecoConv_88201448390851
// MI455X (gfx1250) — compile-verified
//
#include <hip/hip_runtime.h>

#define N_NODES 100000
#define D_CH    64
#define EPS_BN  1e-5f

typedef __attribute__((ext_vector_type(2))) float v2f;
typedef __attribute__((ext_vector_type(8))) float v8f;

// ---------------- K0: zero workspace (h0 accumulator + stats) ----------------
__global__ __launch_bounds__(256) void zero_ws_kernel(float* __restrict__ p, long long n) {
    long long i = (long long)blockIdx.x * blockDim.x + threadIdx.x;
    long long stride = (long long)gridDim.x * blockDim.x;
    for (; i < n; i += stride) p[i] = 0.0f;
}

// ---------------- K1: COO SpMM scatter: h0 = segsum(adj_val * x[col]) --------
// 16 lanes per edge, 4 channels (float4) per lane.
__global__ __launch_bounds__(256) void spmm_kernel(
    const float* __restrict__ x, const float* __restrict__ adj_val,
    const int* __restrict__ adj_row, const int* __restrict__ adj_col,
    float* __restrict__ h0, int E)
{
    long long gid = (long long)blockIdx.x * blockDim.x + threadIdx.x;
    int e = (int)(gid >> 4);
    if (e >= E) return;
    int c4  = ((int)gid & 15) * 4;
    float v = adj_val[e];
    int col = adj_col[e];
    int row = adj_row[e];
    float4 xv = *(const float4*)(x + (size_t)col * D_CH + c4);
    float* dst = h0 + (size_t)row * D_CH + c4;
    atomicAdd(dst + 0, v * xv.x);
    atomicAdd(dst + 1, v * xv.y);
    atomicAdd(dst + 2, v * xv.z);
    atomicAdd(dst + 3, v * xv.w);
}

// ---------------- K2: y = h0 @ W^T + b + x  (fp32 WMMA) + BN stat atomics ----
// One wave handles a 16-row x 64-col strip: 4 C-frags, 16 K-steps of K=4.
__global__ __launch_bounds__(256) void gemm_res_stats_kernel(
    const float* __restrict__ h0, const float* __restrict__ W,
    const float* __restrict__ bias, const float* __restrict__ x,
    float* __restrict__ y, float* __restrict__ ssum, float* __restrict__ ssq)
{
    const int wave = threadIdx.x >> 5;            // 0..7
    const int lane = threadIdx.x & 31;
    const int m0 = (blockIdx.x * 8 + wave) * 16;  // strip base row
    if (m0 >= N_NODES) return;                    // wave-uniform; EXEC stays all-1s

    const int halfsel = lane >> 4;                // 0: lanes 0-15, 1: lanes 16-31
    const int lmod    = lane & 15;
    const int koff    = halfsel * 2;              // A/B K sub-offset per half-wave
    const int arow    = m0 + lmod;

    v8f acc[4];
    acc[0] = (v8f){}; acc[1] = (v8f){}; acc[2] = (v8f){}; acc[3] = (v8f){};

    #pragma unroll
    for (int ks = 0; ks < 16; ++ks) {
        const int kb = ks * 4 + koff;
        // A frag: h0[m0+lmod][kb], h0[m0+lmod][kb+1]
        v2f a = *(const v2f*)(h0 + (size_t)arow * D_CH + kb);
        #pragma unroll
        for (int nt = 0; nt < 4; ++nt) {
            const int n = nt * 16 + lmod;
            // B frag: B[k][n] = W[n][k] -> contiguous float2 of W row n
            v2f bf = *(const v2f*)(W + (size_t)n * D_CH + kb);
            acc[nt] = __builtin_amdgcn_wmma_f32_16x16x4_f32(
                false, a, false, bf, (short)0, acc[nt], false, false);
        }
    }

    // Epilogue: bias + residual, store y, accumulate per-channel sum/sumsq.
    #pragma unroll
    for (int nt = 0; nt < 4; ++nt) {
        const int n = nt * 16 + lmod;
        const float bv = bias[n];
        float s = 0.0f, s2 = 0.0f;
        #pragma unroll
        for (int v = 0; v < 8; ++v) {
            const int row = m0 + v + halfsel * 8;   // C layout: VGPR v -> row v / v+8
            float val = acc[nt][v] + bv + x[(size_t)row * D_CH + n];
            y[(size_t)row * D_CH + n] = val;
            s  += val;
            s2 += val * val;
        }
        atomicAdd(&ssum[n], s);
        atomicAdd(&ssq[n],  s2);
    }
}

// ---------------- K3: fold BN stats into per-channel scale/shift -------------
__global__ void bn_finalize_kernel(const float* __restrict__ ssum,
                                   const float* __restrict__ ssq,
                                   const float* __restrict__ gamma,
                                   const float* __restrict__ beta,
                                   float* __restrict__ scale_shift)
{
    int c = threadIdx.x;
    if (c < D_CH) {
        const float invN = 1.0f / (float)N_NODES;
        float mean = ssum[c] * invN;
        float var  = ssq[c] * invN - mean * mean;
        float inv  = rsqrtf(var + EPS_BN);
        float sc   = gamma[c] * inv;
        scale_shift[c]        = sc;
        scale_shift[D_CH + c] = beta[c] - mean * sc;
    }
}

// ---------------- K4: in-place BN apply + ReLU (float4) ----------------------
__global__ __launch_bounds__(256) void bn_relu_kernel(float* __restrict__ y,
                                                      const float* __restrict__ scale_shift)
{
    long long q = (long long)blockIdx.x * blockDim.x + threadIdx.x;
    long long base = q * 4;
    if (base >= (long long)N_NODES * D_CH) return;
    int c4 = (int)(base & (D_CH - 1));
    float4 v = *(float4*)(y + base);
    const float* sc = scale_shift;
    const float* sh = scale_shift + D_CH;
    v.x = fmaxf(v.x * sc[c4 + 0] + sh[c4 + 0], 0.0f);
    v.y = fmaxf(v.y * sc[c4 + 1] + sh[c4 + 1], 0.0f);
    v.z = fmaxf(v.z * sc[c4 + 2] + sh[c4 + 2], 0.0f);
    v.w = fmaxf(v.w * sc[c4 + 3] + sh[c4 + 3], 0.0f);
    *(float4*)(y + base) = v;
}

extern "C" void kernel_launch(void* const* d_in, const int* in_sizes, int n_in,
                              void* d_out, int out_size, void* d_ws, size_t ws_size,
                              hipStream_t stream) {
    const float* x       = (const float*)d_in[0];
    const float* adj_val = (const float*)d_in[1];
    const float* W       = (const float*)d_in[2];
    const float* b       = (const float*)d_in[3];
    const float* gamma   = (const float*)d_in[4];
    const float* beta    = (const float*)d_in[5];
    const int*   adj_row = (const int*)d_in[6];
    const int*   adj_col = (const int*)d_in[7];
    float* out = (float*)d_out;

    const int E = in_sizes[1];

    // Workspace layout: h0[N*D] | ssum[64] | ssq[64] | scale_shift[128]
    float* h0          = (float*)d_ws;
    float* ssum        = h0 + (size_t)N_NODES * D_CH;
    float* ssq         = ssum + D_CH;
    float* scale_shift = ssq + D_CH;

    // K0: zero h0 + stats (graph-replay determinism: ws is not re-poisoned)
    long long zn = (long long)N_NODES * D_CH + 4 * D_CH;
    zero_ws_kernel<<<2048, 256, 0, stream>>>(h0, zn);

    // K1: SpMM scatter
    long long spmm_threads = (long long)E * 16;
    int spmm_blocks = (int)((spmm_threads + 255) / 256);
    spmm_kernel<<<spmm_blocks, 256, 0, stream>>>(x, adj_val, adj_row, adj_col, h0, E);

    // K2: WMMA GEMM + bias + residual + stats (8 waves/block, 16 rows/wave)
    int strips = (N_NODES + 15) / 16;             // 6250
    int gemm_blocks = (strips + 7) / 8;           // 782
    gemm_res_stats_kernel<<<gemm_blocks, 256, 0, stream>>>(h0, W, b, x, out, ssum, ssq);

    // K3: fold stats
    bn_finalize_kernel<<<1, 64, 0, stream>>>(ssum, ssq, gamma, beta, scale_shift);

    // K4: BN apply + ReLU in place on d_out
    long long quads = (long long)N_NODES * D_CH / 4;
    int bn_blocks = (int)((quads + 255) / 256);
    bn_relu_kernel<<<bn_blocks, 256, 0, stream>>>(out, scale_shift);
}